// MolOptDecoder_65034394796330
// MI455X (gfx1250) — compile-verified
//
#include <hip/hip_runtime.h>

typedef float v2f __attribute__((ext_vector_type(2)));
typedef float v8f __attribute__((ext_vector_type(8)));

namespace {
constexpr int NMOL = 128;                       // B
constexpr int LATM = 40;                        // L
constexpr int ROWS = NMOL * LATM;               // 5120 atoms total
constexpr int NSY  = 40;                        // n symbols
constexpr int NCH  = 5;                         // n charges
constexpr int NBD  = 5;                         // n bond types

constexpr int CHG_OFF = ROWS * NSY;             // 204800
constexpr int BND_OFF = CHG_OFF + ROWS * NCH;   // 230400

// workspace layout in floats
constexpr int ABUF_OFF = 0;                     // [5120][512] Wa@y + b1b
constexpr int BBUF_OFF = ABUF_OFF + ROWS * 512; // [5120][512] Wb@y
constexpr int W2S_OFF  = BBUF_OFF + ROWS * 512; // padded [48][256]
constexpr int W2C_OFF  = W2S_OFF + 48 * 256;    // padded [16][256]
constexpr int W2B_OFF  = W2C_OFF + 16 * 256;    // padded [16][512]

constexpr int YP = 264;                         // LDS row pitch (floats)
}

__device__ __forceinline__ float lrelu(float v) { return v > 0.f ? v : 0.01f * v; }

__device__ __forceinline__ v8f wmma_f32(v2f a, v2f b, v8f c) {
  // D = A(16x4 f32) * B(4x16 f32) + C(16x16 f32)
  return __builtin_amdgcn_wmma_f32_16x16x4_f32(false, a, false, b, (short)0, c,
                                               false, false);
}

// ---------------- kernel 0: build zero-padded second-layer weights ----------
__global__ void init_pads(const float* __restrict__ W2s,
                          const float* __restrict__ W2c,
                          const float* __restrict__ W2b,
                          float* __restrict__ ws) {
  int idx = blockIdx.x * 256 + threadIdx.x;     // grid covers 12288 threads
  if (idx < 48 * 256) {                         // W2s pad: 48 x 256
    int n = idx >> 8, k = idx & 255;
    ws[W2S_OFF + idx] = (n < NSY) ? W2s[n * 256 + k] : 0.f;
  }
  if (idx < 16 * 256) {                         // W2c pad: 16 x 256
    int n = idx >> 8, k = idx & 255;
    ws[W2C_OFF + idx] = (n < NCH) ? W2c[n * 256 + k] : 0.f;
  }
  if (idx < 16 * 512) {                         // W2b pad: 16 x 512
    int n = idx >> 9, k = idx & 511;
    ws[W2B_OFF + idx] = (n < NBD) ? W2b[n * 512 + k] : 0.f;
  }
}

// ---------------- kernel 1: per-atom heads + bond projections ---------------
// One block per 16 atoms. 8 waves. Phase A: y-tile -> {hS, hC, Aproj, Bproj}
// via v_wmma_f32_16x16x4_f32 (96 N-tiles, K=256). Phase B: symbols/charges.
__global__ __launch_bounds__(256) void heads_kernel(
    const float* __restrict__ x,
    const float* __restrict__ W1s, const float* __restrict__ b1s,
    const float* __restrict__ b2s,
    const float* __restrict__ W1c, const float* __restrict__ b1c,
    const float* __restrict__ b2c,
    const float* __restrict__ W1b, const float* __restrict__ b1b,
    float* __restrict__ out, float* __restrict__ ws) {
  __shared__ __align__(16) float yt[16 * YP];
  __shared__ __align__(16) float hS[16 * YP];
  __shared__ __align__(16) float hC[16 * YP];

  const int tid  = threadIdx.x;
  const int lane = tid & 31;
  const int wave = tid >> 5;
  const int r0   = blockIdx.x * 16;             // first atom row of this tile

  // stage 16x256 y-tile into LDS
  for (int e = tid; e < 16 * 256; e += 256) {
    int row = e >> 8, k = e & 255;
    yt[row * YP + k] = x[(r0 + row) * 256 + k];
  }
  __syncthreads();

  const int nl   = lane & 15;                   // row / col within fragment
  const int hi   = lane >> 4;                   // K half select
  const int kadd = hi * 2;

  float* Abuf = ws + ABUF_OFF;
  float* Bbuf = ws + BBUF_OFF;

  // 96 N-tile tasks: [0,16) hS | [16,32) hC | [32,64) Aproj | [64,96) Bproj
  for (int t = wave; t < 96; t += 8) {
    const float* Wp;
    int tile, mode;
    if (t < 16)      { mode = 0; tile = t;      Wp = W1s + (tile * 16 + nl) * 256 + kadd; }
    else if (t < 32) { mode = 1; tile = t - 16; Wp = W1c + (tile * 16 + nl) * 256 + kadd; }
    else if (t < 64) { mode = 2; tile = t - 32; Wp = W1b + (tile * 16 + nl) * 512 + kadd; }
    else             { mode = 3; tile = t - 64; Wp = W1b + (tile * 16 + nl) * 512 + 256 + kadd; }
    const float* Ap = &yt[nl * YP + kadd];

    v8f acc = {};
    #pragma unroll 8
    for (int kk = 0; kk < 64; ++kk) {           // K = 256 in steps of 4
      v2f a = *(const v2f*)(Ap + kk * 4);
      v2f b = *(const v2f*)(Wp + kk * 4);
      acc = wmma_f32(a, b, acc);
    }

    const int ch = tile * 16 + nl;              // output channel
    if (mode == 0) {
      float bias = b1s[ch];
      #pragma unroll
      for (int v = 0; v < 8; ++v) { int row = v + 8 * hi; hS[row * YP + ch] = lrelu(acc[v] + bias); }
    } else if (mode == 1) {
      float bias = b1c[ch];
      #pragma unroll
      for (int v = 0; v < 8; ++v) { int row = v + 8 * hi; hC[row * YP + ch] = lrelu(acc[v] + bias); }
    } else if (mode == 2) {
      float bias = b1b[ch];                     // fold b1b into A projection
      #pragma unroll
      for (int v = 0; v < 8; ++v) { int row = v + 8 * hi; Abuf[(r0 + row) * 512 + ch] = acc[v] + bias; }
    } else {
      #pragma unroll
      for (int v = 0; v < 8; ++v) { int row = v + 8 * hi; Bbuf[(r0 + row) * 512 + ch] = acc[v]; }
    }
  }
  __syncthreads();

  // Phase B: tiny second GEMMs (symbols: 3 N-tiles, charges: 1 N-tile)
  if (wave < 3) {                               // symbols tile = wave
    const float* Wp = ws + W2S_OFF + (wave * 16 + nl) * 256 + kadd;
    const float* Ap = &hS[nl * YP + kadd];
    v8f acc = {};
    #pragma unroll 8
    for (int kk = 0; kk < 64; ++kk) {
      v2f a = *(const v2f*)(Ap + kk * 4);
      v2f b = *(const v2f*)(Wp + kk * 4);
      acc = wmma_f32(a, b, acc);
    }
    int n = wave * 16 + nl;
    if (n < NSY) {
      float bias = b2s[n];
      #pragma unroll
      for (int v = 0; v < 8; ++v) out[(r0 + v + 8 * hi) * NSY + n] = acc[v] + bias;
    }
  } else if (wave == 3) {                       // charges (single tile)
    const float* Wp = ws + W2C_OFF + nl * 256 + kadd;
    const float* Ap = &hC[nl * YP + kadd];
    v8f acc = {};
    #pragma unroll 8
    for (int kk = 0; kk < 64; ++kk) {
      v2f a = *(const v2f*)(Ap + kk * 4);
      v2f b = *(const v2f*)(Wp + kk * 4);
      acc = wmma_f32(a, b, acc);
    }
    int n = nl;
    if (n < NCH) {
      float bias = b2c[n];
      #pragma unroll
      for (int v = 0; v < 8; ++v) out[CHG_OFF + (r0 + v + 8 * hi) * NCH + n] = acc[v] + bias;
    }
  }
}

// ---------------- kernel 2: pairwise bond logits ----------------------------
// One wave per (mol b, atom i, j-tile). Builds lrelu(A_i+B_j) / lrelu(A_j+B_i)
// fragments on the fly and accumulates both orientations with fp32 WMMA.
__global__ __launch_bounds__(32) void bonds_kernel(const float* __restrict__ b2b,
                                                   float* __restrict__ out,
                                                   const float* __restrict__ ws) {
  const int blk  = blockIdx.x;
  const int jt   = blk % 3;                     // j tiles: 0..2 (j in [0,48) masked to 40)
  const int rest = blk / 3;
  const int i    = rest % LATM;
  const int b    = rest / LATM;

  const int lane = threadIdx.x;
  const int nl   = lane & 15;
  const int hi   = lane >> 4;
  const int kadd = hi * 2;

  const float* Abuf = ws + ABUF_OFF;
  const float* Bbuf = ws + BBUF_OFF;

  __shared__ __align__(16) float sAi[512];
  __shared__ __align__(16) float sBi[512];
  const int rowi = b * LATM + i;
  for (int e = lane; e < 512; e += 32) {
    sAi[e] = Abuf[rowi * 512 + e];
    sBi[e] = Bbuf[rowi * 512 + e];
  }
  __syncthreads();

  const int j0 = jt * 16;
  int jrow = j0 + nl; if (jrow > LATM - 1) jrow = LATM - 1;   // clamp pad rows
  const float* pBj = Bbuf + (b * LATM + jrow) * 512 + kadd;
  const float* pAj = Abuf + (b * LATM + jrow) * 512 + kadd;
  const float* pW  = ws + W2B_OFF + nl * 512 + kadd;          // padded W2b col nl
  const float* pAi = sAi + kadd;
  const float* pBi = sBi + kadd;

  v8f acc1 = {};                                // f(i, j0+m)
  v8f acc2 = {};                                // f(j0+m, i)
  #pragma unroll 4
  for (int kk = 0; kk < 128; ++kk) {            // K = 512 in steps of 4
    int k4 = kk * 4;
    v2f ai = *(const v2f*)(pAi + k4);
    v2f bi = *(const v2f*)(pBi + k4);
    v2f bj = *(const v2f*)(pBj + k4);
    v2f aj = *(const v2f*)(pAj + k4);
    v2f w  = *(const v2f*)(pW + k4);
    v2f h1, h2;
    h1.x = lrelu(ai.x + bj.x);  h1.y = lrelu(ai.y + bj.y);
    h2.x = lrelu(aj.x + bi.x);  h2.y = lrelu(aj.y + bi.y);
    acc1 = wmma_f32(h1, w, acc1);
    acc2 = wmma_f32(h2, w, acc2);
  }

  const int n = nl;                             // bond-type channel
  if (n < NBD) {
    float bias2 = 2.f * b2b[n];                 // b2b counted twice by symmetrize
    #pragma unroll
    for (int v = 0; v < 8; ++v) {
      int j = j0 + v + 8 * hi;
      if (j < LATM) {
        float val = (i == j) ? ((n == 4) ? 1000.f : -1000.f)
                             : (acc1[v] + acc2[v] + bias2);
        out[BND_OFF + ((b * LATM + i) * LATM + j) * NBD + n] = val;
      }
    }
  }
}

extern "C" void kernel_launch(void* const* d_in, const int* in_sizes, int n_in,
                              void* d_out, int out_size, void* d_ws, size_t ws_size,
                              hipStream_t stream) {
  const float* x   = (const float*)d_in[0];
  const float* W1s = (const float*)d_in[1];
  const float* b1s = (const float*)d_in[2];
  const float* W2s = (const float*)d_in[3];
  const float* b2s = (const float*)d_in[4];
  const float* W1c = (const float*)d_in[5];
  const float* b1c = (const float*)d_in[6];
  const float* W2c = (const float*)d_in[7];
  const float* b2c = (const float*)d_in[8];
  const float* W1b = (const float*)d_in[9];
  const float* b1b = (const float*)d_in[10];
  const float* W2b = (const float*)d_in[11];
  const float* b2b = (const float*)d_in[12];
  float* out = (float*)d_out;
  float* ws  = (float*)d_ws;

  init_pads<<<48, 256, 0, stream>>>(W2s, W2c, W2b, ws);
  heads_kernel<<<ROWS / 16, 256, 0, stream>>>(x, W1s, b1s, b2s, W1c, b1c, b2c,
                                              W1b, b1b, out, ws);
  bonds_kernel<<<NMOL * LATM * 3, 32, 0, stream>>>(b2b, out, ws);
}